// DeformableAttention_82016695484779
// MI455X (gfx1250) — compile-verified
//
#include <hip/hip_runtime.h>
#include <hip/hip_bf16.h>
#include <math.h>

// ---------------------------------------------------------------------------
// Deformable attention for MI455X (gfx1250, wave32).
// GEMMs use V_WMMA_F32_16X16X4_F32 (full fp32 precision; problem is HBM/L2
// bound at ~7us so fp32 matrix throughput is more than sufficient).
// ---------------------------------------------------------------------------

typedef __attribute__((ext_vector_type(2))) float v2f;
typedef __attribute__((ext_vector_type(8))) float v8f;

#define NUM_B   8
#define NUM_Q   900
#define DIM     256
#define IMG_H   100
#define IMG_W   100
#define HW      (IMG_H * IMG_W)      // 10000
#define N_HEAD  8
#define N_PT    4
#define N_PTS   (N_HEAD * N_PT)      // 32
#define QCOLS   96                   // 64 offset cols + 32 attn cols
#define ROWS_Q  (NUM_B * NUM_Q)      // 7200

// WMMA 16x16x4 f32 fragment helpers (ISA 7.12.2):
//  A (16x4, MxK): lanes 0-15 -> M=lane, VGPR0=K0,VGPR1=K1 ; lanes 16-31 -> K2,K3
//  B (4x16, KxN): lanes 0-15 -> N=lane, VGPR0=K0,VGPR1=K1 ; lanes 16-31 -> K2,K3
//  C/D (16x16):  VGPR j : lanes 0-15 -> M=j,N=lane ; lanes 16-31 -> M=j+8

// ---------------------------------------------------------------------------
// Kernel 1: v[b,hw,e] = sum_d value[b,d,hw] * W_v[d,e] + b_v[e]
// grid(625, 8), block(256) = 8 waves; wave w -> cols [w*32, w*32+32)
// ---------------------------------------------------------------------------
__global__ __launch_bounds__(256)
void deform_vproj(const float* __restrict__ value,
                  const float* __restrict__ Wv,
                  const float* __restrict__ bv,
                  float* __restrict__ vws) {
    const int lane = threadIdx.x & 31;
    const int wave = threadIdx.x >> 5;
    const int half = lane >> 4;      // 0 or 1
    const int l16  = lane & 15;
    const int b    = blockIdx.y;
    const int row0 = blockIdx.x * 16;            // hw tile base (exact: 625*16)
    const int col0 = wave * 32;                  // e tile base

    const float* __restrict__ valB = value + (size_t)b * DIM * HW;
    const int m = row0 + l16;                    // this lane's hw row for A

    v8f acc0 = {};
    v8f acc1 = {};

#pragma unroll 4
    for (int k = 0; k < DIM; k += 4) {
        const int ka = k + 2 * half;
        v2f a, bm0, bm1;
        a.x   = valB[(size_t)ka * HW + m];
        a.y   = valB[(size_t)(ka + 1) * HW + m];
        bm0.x = Wv[ka * DIM + col0 + l16];
        bm0.y = Wv[(ka + 1) * DIM + col0 + l16];
        bm1.x = Wv[ka * DIM + col0 + 16 + l16];
        bm1.y = Wv[(ka + 1) * DIM + col0 + 16 + l16];
        acc0 = __builtin_amdgcn_wmma_f32_16x16x4_f32(false, a, false, bm0,
                                                     (short)0, acc0, false, false);
        acc1 = __builtin_amdgcn_wmma_f32_16x16x4_f32(false, a, false, bm1,
                                                     (short)0, acc1, false, false);
    }

    const float bias0 = bv[col0 + l16];
    const float bias1 = bv[col0 + 16 + l16];
    float* __restrict__ outB = vws + ((size_t)b * HW + row0) * DIM;
#pragma unroll
    for (int j = 0; j < 8; ++j) {
        const int r = j + 8 * half;
        outB[(size_t)r * DIM + col0 + l16]      = acc0[j] + bias0;
        outB[(size_t)r * DIM + col0 + 16 + l16] = acc1[j] + bias1;
    }
}

// ---------------------------------------------------------------------------
// Kernel 2: qp[r, 0:64]  = query[r,:] @ W_off  + b_off
//           qp[r, 64:96] = query[r,:] @ W_attn + b_attn
// grid(450), block(192) = 6 waves; wave w -> cols [w*16, w*16+16)
// ---------------------------------------------------------------------------
__global__ __launch_bounds__(192)
void deform_qproj(const float* __restrict__ query,
                  const float* __restrict__ Woff,
                  const float* __restrict__ boff,
                  const float* __restrict__ Wattn,
                  const float* __restrict__ battn,
                  float* __restrict__ qp) {
    const int lane = threadIdx.x & 31;
    const int wave = threadIdx.x >> 5;
    const int half = lane >> 4;
    const int l16  = lane & 15;
    const int row0 = blockIdx.x * 16;            // 450*16 = 7200 exact
    const int col0 = wave * 16;

    // wave-uniform selection of source matrix (EXEC stays all-ones)
    const bool isOff = (col0 < 64);
    const float* __restrict__ Bp = isOff ? Woff : Wattn;
    const int ldb = isOff ? (N_PTS * 2) : N_PTS; // 64 : 32
    const int cb  = isOff ? (col0 + l16) : (col0 - 64 + l16);

    const int m = row0 + l16;
    v8f acc = {};

#pragma unroll 4
    for (int k = 0; k < DIM; k += 4) {
        const int ka = k + 2 * half;
        v2f a, bm;
        a.x  = query[(size_t)m * DIM + ka];
        a.y  = query[(size_t)m * DIM + ka + 1];
        bm.x = Bp[ka * ldb + cb];
        bm.y = Bp[(ka + 1) * ldb + cb];
        acc = __builtin_amdgcn_wmma_f32_16x16x4_f32(false, a, false, bm,
                                                    (short)0, acc, false, false);
    }

    const float bias = isOff ? boff[cb] : battn[cb];
#pragma unroll
    for (int j = 0; j < 8; ++j) {
        const int r = row0 + j + 8 * half;
        qp[(size_t)r * QCOLS + col0 + l16] = acc[j] + bias;
    }
}

// ---------------------------------------------------------------------------
// Kernel 3: per-query softmax + bilinear gather-accumulate.
// grid(7200), block(256). Lanes 0..31 compute one sampling point each
// (corner weights premultiplied by attn/nH), then all 256 threads gather
// channel-coalesced from vws[b, idx, e].
// ---------------------------------------------------------------------------
__global__ __launch_bounds__(256)
void deform_sample(const float* __restrict__ qp,
                   const float* __restrict__ refpts,
                   const float* __restrict__ vws,
                   float* __restrict__ tmp) {
    __shared__ float s_w[N_PTS][4];
    __shared__ int   s_i[N_PTS][4];

    const int r = blockIdx.x;            // flat (b*900 + q)
    const int b = r / NUM_Q;
    const int tid = threadIdx.x;

    if (tid < N_PTS) {
        const int p  = tid;
        const int h  = p >> 2;
        const int pp = p & 3;

        // softmax over the head's 4 attn logits (bias already fused in qproj)
        const float* lrow = qp + (size_t)r * QCOLS + 64 + h * 4;
        float l0 = lrow[0], l1 = lrow[1], l2 = lrow[2], l3 = lrow[3];
        float mx = fmaxf(fmaxf(l0, l1), fmaxf(l2, l3));
        float e0 = __expf(l0 - mx), e1 = __expf(l1 - mx);
        float e2 = __expf(l2 - mx), e3 = __expf(l3 - mx);
        float inv = 1.0f / (e0 + e1 + e2 + e3);
        float my = (pp == 0) ? e0 : (pp == 1) ? e1 : (pp == 2) ? e2 : e3;
        const float coef = my * inv * (1.0f / (float)N_HEAD);

        const float ox = qp[(size_t)r * QCOLS + h * 8 + pp * 2 + 0];
        const float oy = qp[(size_t)r * QCOLS + h * 8 + pp * 2 + 1];
        const float rx = refpts[(size_t)r * 2 + 0];
        const float ry = refpts[(size_t)r * 2 + 1];

        const float lx = fminf(fmaxf(rx + ox * 0.1f, 0.0f), 1.0f);
        const float ly = fminf(fmaxf(ry + oy * 0.1f, 0.0f), 1.0f);
        const float gx = lx * 2.0f - 1.0f;
        const float gy = ly * 2.0f - 1.0f;
        const float ix = ((gx + 1.0f) * (float)IMG_W - 1.0f) * 0.5f;
        const float iy = ((gy + 1.0f) * (float)IMG_H - 1.0f) * 0.5f;
        const float x0f = floorf(ix), y0f = floorf(iy);
        const float wx1 = ix - x0f, wx0 = 1.0f - wx1;
        const float wy1 = iy - y0f, wy0 = 1.0f - wy1;
        const int x0 = (int)x0f, y0 = (int)y0f;
        const int x1 = x0 + 1,   y1 = y0 + 1;

        #define CORNER(slot, xi, yi, w)                                          \
        {                                                                        \
            const bool valid = ((xi) >= 0) & ((xi) < IMG_W) &                    \
                               ((yi) >= 0) & ((yi) < IMG_H);                     \
            const int cx = min(max((xi), 0), IMG_W - 1);                         \
            const int cy = min(max((yi), 0), IMG_H - 1);                         \
            s_w[p][slot] = valid ? (w) * coef : 0.0f;                            \
            s_i[p][slot] = cy * IMG_W + cx;                                      \
        }
        CORNER(0, x0, y0, wx0 * wy0)
        CORNER(1, x1, y0, wx1 * wy0)
        CORNER(2, x0, y1, wx0 * wy1)
        CORNER(3, x1, y1, wx1 * wy1)
        #undef CORNER
    }
    __syncthreads();

    const int e = tid;
    const float* __restrict__ vb = vws + (size_t)b * HW * DIM;
    float acc = 0.0f;
#pragma unroll 4
    for (int p = 0; p < N_PTS; ++p) {
        for (int c = 0; c < 4; ++c) {
            const float w = s_w[p][c];         // block-uniform -> uniform branch
            if (w != 0.0f)
                acc += w * vb[(size_t)s_i[p][c] * DIM + e];
        }
    }
    tmp[(size_t)r * DIM + e] = acc;
}

// ---------------------------------------------------------------------------
// Kernel 4: out = tmp @ W_out + b_out
// grid(450), block(256) = 8 waves; wave w -> cols [w*32, w*32+32)
// ---------------------------------------------------------------------------
__global__ __launch_bounds__(256)
void deform_outproj(const float* __restrict__ tmp,
                    const float* __restrict__ Wout,
                    const float* __restrict__ bout,
                    float* __restrict__ out) {
    const int lane = threadIdx.x & 31;
    const int wave = threadIdx.x >> 5;
    const int half = lane >> 4;
    const int l16  = lane & 15;
    const int row0 = blockIdx.x * 16;
    const int col0 = wave * 32;

    const int m = row0 + l16;
    v8f acc0 = {};
    v8f acc1 = {};

#pragma unroll 4
    for (int k = 0; k < DIM; k += 4) {
        const int ka = k + 2 * half;
        v2f a, bm0, bm1;
        a.x   = tmp[(size_t)m * DIM + ka];
        a.y   = tmp[(size_t)m * DIM + ka + 1];
        bm0.x = Wout[ka * DIM + col0 + l16];
        bm0.y = Wout[(ka + 1) * DIM + col0 + l16];
        bm1.x = Wout[ka * DIM + col0 + 16 + l16];
        bm1.y = Wout[(ka + 1) * DIM + col0 + 16 + l16];
        acc0 = __builtin_amdgcn_wmma_f32_16x16x4_f32(false, a, false, bm0,
                                                     (short)0, acc0, false, false);
        acc1 = __builtin_amdgcn_wmma_f32_16x16x4_f32(false, a, false, bm1,
                                                     (short)0, acc1, false, false);
    }

    const float bias0 = bout[col0 + l16];
    const float bias1 = bout[col0 + 16 + l16];
#pragma unroll
    for (int j = 0; j < 8; ++j) {
        const int r = row0 + j + 8 * half;
        out[(size_t)r * DIM + col0 + l16]      = acc0[j] + bias0;
        out[(size_t)r * DIM + col0 + 16 + l16] = acc1[j] + bias1;
    }
}

// ---------------------------------------------------------------------------
// Launch. Inputs (setup_inputs order):
//  0 query (8,900,256) f32       1 reference_points (8,900,2) f32
//  2 value (8,256,100,100) f32   3 W_off (256,64)   4 b_off (64)
//  5 W_attn (256,32)             6 b_attn (32)
//  7 W_v (256,256)               8 b_v (256)
//  9 W_out (256,256)            10 b_out (256)
// Output: (8,900,256) f32.
// Workspace: v (20.48M f32) + qproj (691.2K f32) + tmp (1.8432M f32) ~= 92 MB.
// ---------------------------------------------------------------------------
extern "C" void kernel_launch(void* const* d_in, const int* in_sizes, int n_in,
                              void* d_out, int out_size, void* d_ws, size_t ws_size,
                              hipStream_t stream) {
    const float* query  = (const float*)d_in[0];
    const float* refpts = (const float*)d_in[1];
    const float* value  = (const float*)d_in[2];
    const float* Woff   = (const float*)d_in[3];
    const float* boff   = (const float*)d_in[4];
    const float* Wattn  = (const float*)d_in[5];
    const float* battn  = (const float*)d_in[6];
    const float* Wv     = (const float*)d_in[7];
    const float* bv     = (const float*)d_in[8];
    const float* Wout   = (const float*)d_in[9];
    const float* bout   = (const float*)d_in[10];
    float* out = (float*)d_out;

    float* vws = (float*)d_ws;                               // 8*10000*256
    float* qp  = vws + (size_t)NUM_B * HW * DIM;             // 7200*96
    float* tmp = qp + (size_t)ROWS_Q * QCOLS;                // 7200*256

    deform_vproj<<<dim3(HW / 16, NUM_B), 256, 0, stream>>>(value, Wv, bv, vws);
    deform_qproj<<<dim3(ROWS_Q / 16), 192, 0, stream>>>(query, Woff, boff,
                                                        Wattn, battn, qp);
    deform_sample<<<dim3(ROWS_Q), 256, 0, stream>>>(qp, refpts, vws, tmp);
    deform_outproj<<<dim3(ROWS_Q / 16), 256, 0, stream>>>(tmp, Wout, bout, out);
}